// voxel_model_1675037245827
// MI455X (gfx1250) — compile-verified
//
#include <hip/hip_runtime.h>
#include <cstdint>
#include <cstddef>

// ---------------------------------------------------------------------------
// voxel multi-resolution trilinear interpolation, MI455X (gfx1250, wave32)
//
// Strategy:
//   Pass 1 (transpose_k): repack both (12,160,160,160) grids into a single
//     interleaved buffer T[x][y][z][grid][c] (24 contiguous floats / voxel)
//     in d_ws, staging through LDS (async global->LDS when available).
//   Pass 2 (interp_t_k): one thread per point; 3 levels x 8 corners, each
//     corner read as 6x b128 from T; results stored with non-temporal hints.
// ---------------------------------------------------------------------------

#define NPTS      1000000
#define W         160
#define W3        (W*W*W)            // 4,096,000
#define CH        12
#define GC        24                 // 2 grids * 12 channels
#define OUT_HALF  36000000ull        // NPTS * 36 floats (offset of out_pre)

typedef float v4f __attribute__((ext_vector_type(4)));

#ifndef ATHENA_TRY_ASYNC
#define ATHENA_TRY_ASYNC 1
#endif

#if ATHENA_TRY_ASYNC && defined(__has_builtin)
#if __has_builtin(__builtin_amdgcn_global_load_async_to_lds_b32)
#define USE_ASYNC_LDS 1
#endif
#endif

typedef __attribute__((address_space(1))) int g_int;
typedef __attribute__((address_space(3))) int l_int;

// ---------------------------------------------------------------------------
// Pass 1: (C,X,Y,Z) x2 grids  ->  T[(x*W+y)*W+z][gc], gc = grid*12 + c
// One block handles a 32-voxel z-run for one (x,y). Reads are lane-coalesced
// 128B per channel; writes are fully coalesced 3 KB per block.
// ---------------------------------------------------------------------------
__global__ __launch_bounds__(256)
void transpose_k(const float* __restrict__ g0, const float* __restrict__ g1,
                 float* __restrict__ T) {
  __shared__ float tile[32 * 25];          // pad 24->25: conflict-free banks
  const int tid  = threadIdx.x;
  const int lane = tid & 31;
  const int wv   = tid >> 5;               // 8 wave32 waves
  const int b    = blockIdx.x;             // 160*160*5 blocks
  const int zc   = b % 5;
  const int y    = (b / 5) % W;
  const int x    = b / (5 * W);
  const size_t plane = ((size_t)(x * W + y)) * W + (size_t)zc * 32;

  for (int gc = wv; gc < GC; gc += 8) {    // exactly 3 iterations per wave
    const float* src = (gc < CH) ? g0 : g1;
    const int    c   = (gc < CH) ? gc : gc - CH;
    const float* gp  = src + (size_t)c * W3 + plane + (size_t)lane;
    const int    li  = lane * 25 + gc;
#if defined(USE_ASYNC_LDS)
    __builtin_amdgcn_global_load_async_to_lds_b32(
        (g_int*)gp, (l_int*)&tile[li], /*offset=*/0, /*cpol=*/0);
#else
    tile[li] = __builtin_nontemporal_load(gp);   // read-once: NT
#endif
  }
#if defined(USE_ASYNC_LDS)
#if __has_builtin(__builtin_amdgcn_s_wait_asynccnt)
  __builtin_amdgcn_s_wait_asynccnt(0);
#else
  asm volatile("s_wait_asynccnt 0" ::: "memory");
#endif
#endif
  __syncthreads();

  float* dst = T + plane * GC;             // 768 contiguous floats
  for (int i = tid; i < 32 * GC; i += 256) {
    const int zl = i / GC;
    const int gc = i - zl * GC;
    dst[i] = tile[zl * 25 + gc];
  }
}

// ---------------------------------------------------------------------------
// Trilinear sample of one mip level from the interleaved buffer.
// Padded level size S in {161,81,41}, stride s in {1,2,4}. Original index
// idx*s == 160 corresponds to the reference's zero-pad layer -> skip (adds 0).
// ---------------------------------------------------------------------------
__device__ __forceinline__
void interp_level(const float* __restrict__ T, float tx, float ty, float tz,
                  int S, int s, v4f acc[6]) {
  const float Sm1 = (float)(S - 1);
  const float px = tx * Sm1, py = ty * Sm1, pz = tz * Sm1;
  int ix = (int)floorf(px), iy = (int)floorf(py), iz = (int)floorf(pz);
  ix = ix < 0 ? 0 : (ix > S - 2 ? S - 2 : ix);
  iy = iy < 0 ? 0 : (iy > S - 2 ? S - 2 : iy);
  iz = iz < 0 ? 0 : (iz > S - 2 ? S - 2 : iz);
  const float fx = px - (float)ix, fy = py - (float)iy, fz = pz - (float)iz;
  const float wx[2] = {1.0f - fx, fx};
  const float wy[2] = {1.0f - fy, fy};
  const float wz[2] = {1.0f - fz, fz};

#pragma unroll
  for (int k = 0; k < 6; ++k) acc[k] = (v4f)0.0f;

#pragma unroll
  for (int dx = 0; dx < 2; ++dx) {
    const int X = (ix + dx) * s;
    if (X >= W) continue;                          // zero-pad layer
#pragma unroll
    for (int dy = 0; dy < 2; ++dy) {
      const int Y = (iy + dy) * s;
      if (Y >= W) continue;
      const float wxy = wx[dx] * wy[dy];
      const size_t rb = ((size_t)X * W + (size_t)Y) * (size_t)W;
#pragma unroll
      for (int dz = 0; dz < 2; ++dz) {
        const int Z = (iz + dz) * s;
        if (Z >= W) continue;
        const float ww = wxy * wz[dz];
        const v4f* cp = (const v4f*)(T + (rb + (size_t)Z) * GC);
#pragma unroll
        for (int k = 0; k < 6; ++k) acc[k] += ww * cp[k];
      }
    }
  }
}

__global__ __launch_bounds__(256)
void interp_t_k(const float* __restrict__ xyz, const float* __restrict__ T,
                float* __restrict__ out) {
  const int p = blockIdx.x * 256 + threadIdx.x;
  if (p >= NPTS) return;
  __builtin_prefetch(xyz + 3 * (size_t)p, 0, 0);   // global_prefetch_b8
  const float x = xyz[3 * p + 0];
  const float y = xyz[3 * p + 1];
  const float z = xyz[3 * p + 2];
  const float tx = (x + 1.5f) / 3.0f;
  const float ty = (y + 1.5f) / 3.0f;
  const float tz = (z + 1.5f) / 3.0f;
  const int Ss[3] = {161, 81, 41};
  const int ss[3] = {1, 2, 4};
#pragma unroll
  for (int l = 0; l < 3; ++l) {
    v4f acc[6];
    interp_level(T, tx, ty, tz, Ss[l], ss[l], acc);
    v4f* o0 = (v4f*)(out + (size_t)p * 36 + (size_t)l * 12);
    v4f* o1 = (v4f*)(out + OUT_HALF + (size_t)p * 36 + (size_t)l * 12);
    // 288 MB of results, never re-read: keep them out of L2 (TH=NT).
    __builtin_nontemporal_store(acc[0], o0 + 0);
    __builtin_nontemporal_store(acc[1], o0 + 1);
    __builtin_nontemporal_store(acc[2], o0 + 2);
    __builtin_nontemporal_store(acc[3], o1 + 0);
    __builtin_nontemporal_store(acc[4], o1 + 1);
    __builtin_nontemporal_store(acc[5], o1 + 2);
  }
}

// ---------------------------------------------------------------------------
// Fallback: gather straight from the (C,X,Y,Z) layout (used only if d_ws is
// too small for the interleaved buffer). Correct but more HBM traffic.
// ---------------------------------------------------------------------------
__device__ __forceinline__
void interp_level_direct(const float* __restrict__ g0, const float* __restrict__ g1,
                         float tx, float ty, float tz, int S, int s, float acc[GC]) {
  const float Sm1 = (float)(S - 1);
  const float px = tx * Sm1, py = ty * Sm1, pz = tz * Sm1;
  int ix = (int)floorf(px), iy = (int)floorf(py), iz = (int)floorf(pz);
  ix = ix < 0 ? 0 : (ix > S - 2 ? S - 2 : ix);
  iy = iy < 0 ? 0 : (iy > S - 2 ? S - 2 : iy);
  iz = iz < 0 ? 0 : (iz > S - 2 ? S - 2 : iz);
  const float fx = px - (float)ix, fy = py - (float)iy, fz = pz - (float)iz;
  const float wx[2] = {1.0f - fx, fx};
  const float wy[2] = {1.0f - fy, fy};
  const float wz[2] = {1.0f - fz, fz};
#pragma unroll
  for (int k = 0; k < GC; ++k) acc[k] = 0.0f;
#pragma unroll
  for (int dx = 0; dx < 2; ++dx) {
    const int X = (ix + dx) * s;
    if (X >= W) continue;
#pragma unroll
    for (int dy = 0; dy < 2; ++dy) {
      const int Y = (iy + dy) * s;
      if (Y >= W) continue;
      const float wxy = wx[dx] * wy[dy];
      const size_t rb = ((size_t)X * W + (size_t)Y) * (size_t)W;
#pragma unroll
      for (int dz = 0; dz < 2; ++dz) {
        const int Z = (iz + dz) * s;
        if (Z >= W) continue;
        const float ww = wxy * wz[dz];
        const size_t vox = rb + (size_t)Z;
#pragma unroll
        for (int c = 0; c < CH; ++c) {
          acc[c]      += ww * g0[(size_t)c * W3 + vox];
          acc[CH + c] += ww * g1[(size_t)c * W3 + vox];
        }
      }
    }
  }
}

__global__ __launch_bounds__(256)
void interp_direct_k(const float* __restrict__ xyz,
                     const float* __restrict__ g0, const float* __restrict__ g1,
                     float* __restrict__ out) {
  const int p = blockIdx.x * 256 + threadIdx.x;
  if (p >= NPTS) return;
  const float x = xyz[3 * p + 0];
  const float y = xyz[3 * p + 1];
  const float z = xyz[3 * p + 2];
  const float tx = (x + 1.5f) / 3.0f;
  const float ty = (y + 1.5f) / 3.0f;
  const float tz = (z + 1.5f) / 3.0f;
  const int Ss[3] = {161, 81, 41};
  const int ss[3] = {1, 2, 4};
#pragma unroll
  for (int l = 0; l < 3; ++l) {
    float acc[GC];
    interp_level_direct(g0, g1, tx, ty, tz, Ss[l], ss[l], acc);
    float* o0 = out + (size_t)p * 36 + (size_t)l * 12;
    float* o1 = out + OUT_HALF + (size_t)p * 36 + (size_t)l * 12;
#pragma unroll
    for (int c = 0; c < CH; ++c) {
      __builtin_nontemporal_store(acc[c],      o0 + c);
      __builtin_nontemporal_store(acc[CH + c], o1 + c);
    }
  }
}

// ---------------------------------------------------------------------------
extern "C" void kernel_launch(void* const* d_in, const int* in_sizes, int n_in,
                              void* d_out, int out_size, void* d_ws, size_t ws_size,
                              hipStream_t stream) {
  (void)in_sizes; (void)n_in; (void)out_size;
  const float* xyz = (const float*)d_in[0];
  const float* k0  = (const float*)d_in[1];
  const float* k0p = (const float*)d_in[2];
  float* out = (float*)d_out;

  const size_t need = (size_t)W3 * GC * sizeof(float);   // ~393 MB
  const int interp_blocks = (NPTS + 255) / 256;

  if (ws_size >= need) {
    float* T = (float*)d_ws;
    transpose_k<<<dim3(W * W * 5), dim3(256), 0, stream>>>(k0, k0p, T);
    interp_t_k<<<dim3(interp_blocks), dim3(256), 0, stream>>>(xyz, T, out);
  } else {
    interp_direct_k<<<dim3(interp_blocks), dim3(256), 0, stream>>>(xyz, k0, k0p, out);
  }
}